// gated_tpp_2010044695333
// MI455X (gfx1250) — compile-verified
//
#include <hip/hip_runtime.h>
#include <math.h>

typedef float v2f __attribute__((ext_vector_type(2)));
typedef float v8f __attribute__((ext_vector_type(8)));

#define B_  8
#define S_  2048
#define D_  64
#define NS_ 50

// ---------------------------------------------------------------------------
// Kernel A: positional embedding, stored PAIR-INTERLEAVED over the sequence
// dimension so kernel B's WMMA B-operand (rows j, j+1 at one column) is a
// single aligned 8-byte load:  pep[b][ (j>>1)*128 + d*2 + (j&1) ].
// Also emits the scalar type embedding te[B,S].
// ---------------------------------------------------------------------------
__global__ void __launch_bounds__(256)
k_encode(const float* __restrict__ event_time,
         const int*   __restrict__ event_type,
         const float* __restrict__ Wt_pos,      // [32,1]
         const float* __restrict__ type_table,  // [11,1]
         float* __restrict__ pep, float* __restrict__ te) {
  int gid = blockIdx.x * blockDim.x + threadIdx.x;   // over B*S*D
  if (gid >= B_ * S_ * D_) return;
  int d  = gid & 63;
  int bs = gid >> 6;
  int s  = bs & (S_ - 1);
  int b  = bs >> 11;
  float t = event_time[bs];
  int k = (d < 32) ? d : d - 32;
  // div_term[k] = exp(2k * (-ln(10000)/64))
  float div = __expf((float)(2 * k) * (-9.210340371976184f / 64.0f));
  float ang = (float)s * div + t * Wt_pos[k];
  float v = (d < 32) ? __sinf(ang) : __cosf(ang);
  pep[b * S_ * D_ + ((s >> 1) * 64 + d) * 2 + (s & 1)] = v;
  if (d == 0) {
    int ty = event_type[bs];
    float e = type_table[ty] * 8.0f;              // * sqrt(D)
    te[bs] = (ty != 0) ? e : 0.0f;                // padding_idx=0
  }
}

// ---------------------------------------------------------------------------
// Kernel B: hidden = causal RQ-kernel scores @ pe, fused with LayerNorm.
// One wave32 per 16-row i-tile; scores built in-register directly into the
// V_WMMA_F32_16X16X4_F32 A-operand layout; fast v_rcp for RQ + sigmoid.
// The causal mask can only trigger in the diagonal tile, so that tile is
// peeled and interior tiles run mask-free.
// ---------------------------------------------------------------------------
__global__ void __launch_bounds__(32)
k_attn(const float* __restrict__ event_time,
       const float* __restrict__ te,
       const float* __restrict__ pep,
       const float* __restrict__ gamma,
       const float* __restrict__ beta,
       float* __restrict__ out_hidden) {
  __shared__ float tile[16 * 64];
  __shared__ float mu_s[16];
  __shared__ float rs_s[16];

  int b    = blockIdx.y;
  int i0   = blockIdx.x * 16;
  int L    = threadIdx.x;
  int m    = L & 15;
  int half = L >> 4;
  int i    = i0 + m;

  const float* t_row = event_time + b * S_;
  const float* e_row = te         + b * S_;
  const float* peb   = pep + b * S_ * D_;

  float ti = t_row[i];
  float di = e_row[i];

  v8f acc[4];
  #pragma unroll
  for (int nt = 0; nt < 4; nt++) acc[nt] = (v8f){0.f,0.f,0.f,0.f,0.f,0.f,0.f,0.f};

  // One j-tile: compute 8 scores per lane (A-operand layout) and issue the
  // 16 WMMAs.  MASKED=true only for the diagonal tile.
  auto do_tile = [&](int j0, bool masked) {
    float sc[8];
    #pragma unroll
    for (int p = 0; p < 8; p++) {
      int c = 4 * (p >> 1) + 2 * half + (p & 1);
      int j = j0 + c;
      float dt   = ti - t_row[j];
      float base = __builtin_amdgcn_rcpf(fmaf(0.5f * dt, dt, 1.0f)); // RQ
      float g    = fabsf(di - e_row[j]);
      float gate = __builtin_amdgcn_rcpf(1.0f + __expf(g));          // sigmoid(-|g|)
      float s = base * gate;
      sc[p] = (masked && j > i) ? 0.0f : s;                          // causal
    }
    #pragma unroll
    for (int k = 0; k < 4; k++) {
      v2f a; a.x = sc[2 * k]; a.y = sc[2 * k + 1];
      int r0 = j0 + 4 * k + 2 * half;          // even: K rows r0, r0+1
      const float* p0 = peb + (r0 >> 1) * 128; // pair-interleaved base
      #pragma unroll
      for (int nt = 0; nt < 4; nt++) {
        int col = nt * 16 + m;
        float2 bv = *(const float2*)(p0 + col * 2);   // rows r0, r0+1
        v2f bb; bb.x = bv.x; bb.y = bv.y;
        acc[nt] = __builtin_amdgcn_wmma_f32_16x16x4_f32(
            false, a, false, bb, (short)0, acc[nt], false, false);
      }
    }
  };

  for (int j0 = 0; j0 < i0; j0 += 16) do_tile(j0, false);  // interior: no mask
  do_tile(i0, true);                                       // diagonal: masked

  // Spill 16x64 tile to LDS (lane L holds rows r+8*half, col nt*16+m).
  #pragma unroll
  for (int nt = 0; nt < 4; nt++)
    #pragma unroll
    for (int r = 0; r < 8; r++)
      tile[(r + 8 * half) * 64 + nt * 16 + m] = acc[nt][r];
  __syncthreads();

  // LayerNorm stats: 16 lanes, one row each.
  if (L < 16) {
    float sum = 0.f, sq = 0.f;
    #pragma unroll
    for (int c = 0; c < 64; c++) {
      float v = tile[L * 64 + c];
      sum += v; sq += v * v;
    }
    float mu  = sum * (1.0f / 64.0f);
    float var = sq * (1.0f / 64.0f) - mu * mu;
    mu_s[L] = mu;
    rs_s[L] = rsqrtf(var + 1e-6f);
  }
  __syncthreads();

  // Coalesced normalized write-out.
  float* orow = out_hidden + (b * S_ + i0) * 64;
  for (int idx = L; idx < 16 * 64; idx += 32) {
    int r = idx >> 6, c = idx & 63;
    orow[idx] = (tile[idx] - mu_s[r]) * rs_s[r] * gamma[c] + beta[c];
  }
}

// ---------------------------------------------------------------------------
// Kernel C: GAN decoder.  Per (b,s): hs = relu(noise @ W_noise^T + proj),
// t = softplus(hs . W_event), mean over 50 samples.  One wave32 per (b,s),
// 8 waves per block; W_noise staged in LDS (ds b64 loads land exactly on the
// B-operand layout).
// ---------------------------------------------------------------------------
__global__ void __launch_bounds__(256)
k_decode(const float* __restrict__ noise,
         const float* __restrict__ W_in,
         const float* __restrict__ W_noise,
         const float* __restrict__ W_event,
         const float* __restrict__ hidden,
         float* __restrict__ out_mean) {
  __shared__ float w_lds[64 * 64];    // W_noise row-major [e][d]
  __shared__ float we_lds[64];
  __shared__ float proj_lds[8][64];

  int t = threadIdx.x;
  for (int idx = t; idx < 64 * 64; idx += 256) w_lds[idx] = W_noise[idx];
  if (t < 64) we_lds[t] = W_event[t];
  __syncthreads();

  int wv   = t >> 5;
  int L    = t & 31;
  int m    = L & 15;
  int half = L >> 4;
  int bs   = blockIdx.x * 8 + wv;     // flattened (b*S + s)

  // proj[e] = hidden[bs,:] . W_in[e,:]  (2 columns per lane)
  const float* hrow = hidden + bs * 64;
  #pragma unroll
  for (int q = 0; q < 2; q++) {
    int e = L + 32 * q;
    const float* wr = W_in + e * 64;
    float a = 0.f;
    for (int d = 0; d < 64; d++) a += hrow[d] * wr[d];
    proj_lds[wv][e] = a;
  }
  __syncthreads();

  const float* nbase = noise + (long)bs * NS_ * 64;
  float total = 0.f;

  for (int mt = 0; mt < 4; mt++) {
    int n = mt * 16 + m;                       // A-operand row (sample index)
    v8f acc[4];
    #pragma unroll
    for (int nt = 0; nt < 4; nt++) acc[nt] = (v8f){0.f,0.f,0.f,0.f,0.f,0.f,0.f,0.f};

    const float* arow = nbase + n * 64;
    #pragma unroll
    for (int kc = 0; kc < 16; kc++) {
      int c0 = 4 * kc + 2 * half;              // A cols / B rows K, K+1
      v2f a;
      if (n < NS_) {
        float2 av = *(const float2*)(arow + c0);
        a.x = av.x; a.y = av.y;
      } else {
        a.x = 0.f; a.y = 0.f;                  // zero-pad rows 50..63
      }
      #pragma unroll
      for (int nt = 0; nt < 4; nt++) {
        float2 wb = *(const float2*)&w_lds[(nt * 16 + m) * 64 + c0]; // B=W_noise^T
        v2f bb; bb.x = wb.x; bb.y = wb.y;
        acc[nt] = __builtin_amdgcn_wmma_f32_16x16x4_f32(
            false, a, false, bb, (short)0, acc[nt], false, false);
      }
    }

    // Epilogue: +proj, relu, dot with W_event along e, reduce across columns.
    float rowacc[8];
    #pragma unroll
    for (int r = 0; r < 8; r++) rowacc[r] = 0.f;
    #pragma unroll
    for (int nt = 0; nt < 4; nt++) {
      float pj = proj_lds[wv][nt * 16 + m];
      float we = we_lds[nt * 16 + m];
      #pragma unroll
      for (int r = 0; r < 8; r++) {
        float v = acc[nt][r] + pj;
        v = fmaxf(v, 0.f);
        rowacc[r] += v * we;
      }
    }
    // Sum over the 16 column-lanes within each half-wave.
    #pragma unroll
    for (int off = 1; off < 16; off <<= 1)
      #pragma unroll
      for (int r = 0; r < 8; r++)
        rowacc[r] += __shfl_xor(rowacc[r], off, 32);

    if (m == 0) {
      #pragma unroll
      for (int r = 0; r < 8; r++) {
        int nn = mt * 16 + r + 8 * half;       // D row -> sample index
        if (nn < NS_) {
          float z = rowacc[r];
          // softplus (stable): large z -> z
          total += (z > 20.f) ? z : __logf(1.0f + __expf(z));
        }
      }
    }
  }
  // Combine lane 0 and lane 16 partials (others carry zero).
  #pragma unroll
  for (int off = 16; off >= 1; off >>= 1)
    total += __shfl_xor(total, off, 32);
  if (L == 0) out_mean[bs] = total * (1.0f / NS_);
}

// ---------------------------------------------------------------------------
extern "C" void kernel_launch(void* const* d_in, const int* in_sizes, int n_in,
                              void* d_out, int out_size, void* d_ws, size_t ws_size,
                              hipStream_t stream) {
  const int*   event_type = (const int*)  d_in[0];
  const float* event_time = (const float*)d_in[1];
  // d_in[2] arrival_times: unused by the forward pass
  const float* noise      = (const float*)d_in[3];
  const float* Wt_pos     = (const float*)d_in[4];
  const float* type_table = (const float*)d_in[5];
  const float* gamma      = (const float*)d_in[6];
  const float* beta       = (const float*)d_in[7];
  const float* W_in       = (const float*)d_in[8];
  const float* W_noise    = (const float*)d_in[9];
  const float* W_event    = (const float*)d_in[10];

  float* out_mean   = (float*)d_out;            // [B,S]
  float* out_hidden = out_mean + B_ * S_;       // [B,S,D]

  float* pep = (float*)d_ws;                    // B*S*D f32 = 4 MB (pair-interleaved)
  float* te  = pep + B_ * S_ * D_;              // B*S   f32 = 64 KB

  k_encode<<<(B_ * S_ * D_ + 255) / 256, 256, 0, stream>>>(
      event_time, event_type, Wt_pos, type_table, pep, te);

  k_attn<<<dim3(S_ / 16, B_), 32, 0, stream>>>(
      event_time, te, pep, gamma, beta, out_hidden);

  k_decode<<<(B_ * S_) / 8, 256, 0, stream>>>(
      noise, W_in, W_noise, W_event, out_hidden, out_mean);
}